// BaggingMaxPool_9277129359373
// MI455X (gfx1250) — compile-verified
//
#include <hip/hip_runtime.h>
#include <math.h>

// Problem constants (from reference)
#define N_ROWS   1024
#define D_COLS   100000
#define K_ROUNDS 20
#define SEL_N    256
#define G4       (D_COLS / 4)      // 25000 float4 column-groups

// Tiling
#define WAVES          8
#define TPB            (WAVES * 32)           // 256 threads
#define ROWS_PER_WAVE  (N_ROWS / WAVES)       // 128
#define SBATCH         8                      // rows per async batch
#define NBATCH         (ROWS_PER_WAVE / SBATCH) // 16
#define COLS_PER_BLK   128                    // 32 lanes * 4 floats

typedef int v4i __attribute__((ext_vector_type(4)));
typedef __attribute__((address_space(1))) v4i* g1v4i_t;   // global int4*
typedef __attribute__((address_space(3))) v4i* l3v4i_t;   // LDS int4*

// ---------------------------------------------------------------------------
// Kernel 1: build per-row 20-bit round-membership masks. 5120 atomicOr in LDS.
// ---------------------------------------------------------------------------
__global__ void __launch_bounds__(N_ROWS)
bmp_build_masks(const int* __restrict__ idx, unsigned* __restrict__ masks) {
  __shared__ unsigned sm[N_ROWS];
  const int t = threadIdx.x;
  sm[t] = 0u;
  __syncthreads();
  for (int i = t; i < K_ROUNDS * SEL_N; i += N_ROWS) {
    const int row = idx[i];             // 0..1023
    atomicOr(&sm[row], 1u << (i / SEL_N));
  }
  __syncthreads();
  masks[t] = sm[t];
}

// ---------------------------------------------------------------------------
// Kernel 2: single pass over the input. Each block = 128 columns; each wave =
// disjoint 128-row strip streamed via double-buffered async global->LDS loads.
// ---------------------------------------------------------------------------
__global__ void __launch_bounds__(TPB)
bmp_main(const float* __restrict__ inp, const unsigned* __restrict__ masks,
         float* __restrict__ out) {
  // 8 waves * 2 buffers * 8 rows * 128 floats = 16384 floats = 64 KB
  __shared__ float    stage[WAVES * 2 * SBATCH * COLS_PER_BLK];
  __shared__ unsigned maskbuf[N_ROWS];   // 4 KB

  const int tid  = threadIdx.x;
  const int wave = tid >> 5;
  const int lane = tid & 31;

  // Stage masks into LDS once per block.
  for (int i = tid; i < N_ROWS; i += TPB) maskbuf[i] = masks[i];
  __syncthreads();

  // Column group for this lane (clamped for the tail block; dup loads are ok,
  // the final store is guarded).
  int g4 = blockIdx.x * 32 + lane;
  if (g4 > G4 - 1) g4 = G4 - 1;

  const int wave_lds = wave * (2 * SBATCH * COLS_PER_BLK); // float index
  const int lane4    = lane * 4;
  const int row0     = wave * ROWS_PER_WAVE;

  float4 mx[K_ROUNDS];
#pragma unroll
  for (int k = 0; k < K_ROUNDS; ++k)
    mx[k] = make_float4(-INFINITY, -INFINITY, -INFINITY, -INFINITY);

  // Issue one batch of SBATCH rows into LDS buffer p via the async copy engine.
  auto issue = [&](int b, int p) {
    const int rb = row0 + b * SBATCH;
#pragma unroll
    for (int s = 0; s < SBATCH; ++s) {
      float* src = const_cast<float*>(inp) + (size_t)(rb + s) * D_COLS
                   + (size_t)g4 * 4;
      float* dst = &stage[wave_lds + p * (SBATCH * COLS_PER_BLK)
                          + s * COLS_PER_BLK + lane4];
      __builtin_amdgcn_global_load_async_to_lds_b128(
          (g1v4i_t)src, (l3v4i_t)dst, /*offset=*/0, /*cpol=*/0);
    }
  };

  issue(0, 0);

#pragma unroll 1
  for (int b = 0; b < NBATCH; ++b) {
    const int p = b & 1;
    if (b + 1 < NBATCH) {
      issue(b + 1, p ^ 1);
      __builtin_amdgcn_s_wait_asynccnt(SBATCH);   // batch b complete
    } else {
      __builtin_amdgcn_s_wait_asynccnt(0);
    }
    asm volatile("" ::: "memory");                // keep LDS reads below wait

    const int rb = row0 + b * SBATCH;
#pragma unroll
    for (int s = 0; s < SBATCH; ++s) {
      const float4 v = *(const float4*)&stage[wave_lds
                          + p * (SBATCH * COLS_PER_BLK) + s * COLS_PER_BLK + lane4];
      // mask is wave-uniform -> scalar branches per round
      const unsigned m =
          (unsigned)__builtin_amdgcn_readfirstlane((int)maskbuf[rb + s]);
#pragma unroll
      for (int k = 0; k < K_ROUNDS; ++k) {
        if (m & (1u << k)) {
          mx[k].x = fmaxf(mx[k].x, v.x);
          mx[k].y = fmaxf(mx[k].y, v.y);
          mx[k].z = fmaxf(mx[k].z, v.z);
          mx[k].w = fmaxf(mx[k].w, v.w);
        }
      }
    }
  }

  // -------- cross-wave reduction in LDS (reuse stage as red[K][128]) --------
  __syncthreads();
  float* red = stage;
  for (int i = tid; i < K_ROUNDS * COLS_PER_BLK; i += TPB) red[i] = -INFINITY;
  __syncthreads();

#pragma unroll 1
  for (int w = 0; w < WAVES; ++w) {
    if (wave == w) {
#pragma unroll
      for (int k = 0; k < K_ROUNDS; ++k) {
        float* r = &red[k * COLS_PER_BLK + lane4];
        r[0] = fmaxf(r[0], mx[k].x);
        r[1] = fmaxf(r[1], mx[k].y);
        r[2] = fmaxf(r[2], mx[k].z);
        r[3] = fmaxf(r[3], mx[k].w);
      }
    }
    __syncthreads();
  }

  // -------- mean over rounds, store --------
  if (tid < COLS_PER_BLK) {
    const int gcol = blockIdx.x * COLS_PER_BLK + tid;
    if (gcol < D_COLS) {
      float sum = 0.0f;
#pragma unroll
      for (int k = 0; k < K_ROUNDS; ++k) sum += red[k * COLS_PER_BLK + tid];
      out[gcol] = sum * (1.0f / (float)K_ROUNDS);
    }
  }
}

// ---------------------------------------------------------------------------
extern "C" void kernel_launch(void* const* d_in, const int* in_sizes, int n_in,
                              void* d_out, int out_size, void* d_ws, size_t ws_size,
                              hipStream_t stream) {
  const float* inp  = (const float*)d_in[0];
  const int*   idx  = (const int*)d_in[1];
  unsigned*    msk  = (unsigned*)d_ws;       // 4 KB scratch
  float*       out  = (float*)d_out;

  bmp_build_masks<<<1, N_ROWS, 0, stream>>>(idx, msk);

  const int nblk = (G4 + 31) / 32;           // 782 blocks of 128 columns
  bmp_main<<<nblk, TPB, 0, stream>>>(inp, msk, out);
}